// StructureEncoder_70866960384624
// MI455X (gfx1250) — compile-verified
//
#include <hip/hip_runtime.h>
#include <hip/hip_bf16.h>

typedef __attribute__((ext_vector_type(2))) float v2f;
typedef __attribute__((ext_vector_type(8))) float v8f;

#define N_NODES  100000
#define N_EDGES  3200000LL
#define N_GRAPHS 256

// ---------------------------------------------------------------------------
// Small utility kernels
// ---------------------------------------------------------------------------
__global__ void fill_kernel(float* __restrict__ p, float v, int n) {
    int i = blockIdx.x * blockDim.x + threadIdx.x;
    if (i < n) p[i] = v;
}

// Convert int64 edge lists to int32 once; accumulate in-degree (deg pre-filled 1.0 for self loop).
__global__ void edge_prep_kernel(const long long* __restrict__ src64,
                                 const long long* __restrict__ dst64,
                                 int* __restrict__ srcI, int* __restrict__ dstI,
                                 float* __restrict__ deg, long long nE) {
    long long e = (long long)blockIdx.x * blockDim.x + threadIdx.x;
    if (e >= nE) return;
    if ((threadIdx.x & 63) == 0) {
        __builtin_prefetch(src64 + e + 4096, 0, 0);  // global_prefetch_b8, streaming hint
        __builtin_prefetch(dst64 + e + 4096, 0, 0);
    }
    int s = (int)src64[e];
    int d = (int)dst64[e];
    srcI[e] = s;
    dstI[e] = d;
    unsafeAtomicAdd(deg + d, 1.0f);   // global_atomic_add_f32
}

__global__ void rsqrt_kernel(float* __restrict__ deg, int n) {
    int i = blockIdx.x * blockDim.x + threadIdx.x;
    if (i < n) deg[i] = rsqrtf(deg[i]);   // deg >= 1 always (self loop)
}

// norm[e] = dinv[src]*dinv[dst]; computed once, reused by all 3 layers.
__global__ void norm_kernel(const int* __restrict__ srcI, const int* __restrict__ dstI,
                            const float* __restrict__ dinv, float* __restrict__ nrm,
                            long long nE) {
    long long e = (long long)blockIdx.x * blockDim.x + threadIdx.x;
    if (e >= nE) return;
    nrm[e] = dinv[srcI[e]] * dinv[dstI[e]];
}

// ---------------------------------------------------------------------------
// Dense GEMM: out[N,KOUT] = A[N,KIN] @ W[KIN,KOUT] using V_WMMA_F32_16X16X4_F32.
// One wave per 16x16 output tile; W staged in LDS (zero-padded K to multiple of 4).
// ---------------------------------------------------------------------------
template <int KIN, int KOUT>
__global__ __launch_bounds__(256) void gemm_wmma_kernel(const float* __restrict__ A,
                                                        const float* __restrict__ W,
                                                        float* __restrict__ out,
                                                        int nrows) {
    constexpr int KP = (KIN + 3) & ~3;           // padded K
    constexpr int colTiles = KOUT / 16;
    __shared__ float sW[KP * KOUT];

    for (int i = threadIdx.x; i < KP * KOUT; i += 256)
        sW[i] = (i < KIN * KOUT) ? W[i] : 0.0f;  // pad rows >= KIN with zeros
    __syncthreads();

    const int wave = threadIdx.x >> 5;
    const int lane = threadIdx.x & 31;
    const long long wg = (long long)blockIdx.x * 8 + wave;
    const long long totalWaves = (long long)(nrows >> 4) * colTiles;
    if (wg >= totalWaves) return;                // wave-uniform exit (EXEC stays all-1s)

    const int tr = (int)(wg / colTiles);
    const int tc = (int)(wg % colTiles);
    const int row  = tr * 16 + (lane & 15);      // A: lane -> M row
    const int colG = tc * 16 + (lane & 15);      // B/D: lane -> N column
    const int kb   = (lane >> 4) << 1;           // lane half -> K pair {0,1} or {2,3}

    v8f c = {};
    for (int k0 = 0; k0 < KP; k0 += 4) {
        const int k = k0 + kb;
        v2f a, b;
        a.x = (KIN % 4 == 0 || k     < KIN) ? A[(long long)row * KIN + k]     : 0.0f;
        a.y = (KIN % 4 == 0 || k + 1 < KIN) ? A[(long long)row * KIN + k + 1] : 0.0f;
        b.x = sW[k * KOUT + colG];
        b.y = sW[(k + 1) * KOUT + colG];
        c = __builtin_amdgcn_wmma_f32_16x16x4_f32(false, a, false, b,
                                                  (short)0, c, false, false);
    }

    // D layout: VGPR j -> M = j (lanes 0-15) / M = 8+j (lanes 16-31), N = lane&15
    const int rbase = tr * 16 + ((lane >> 4) << 3);
#pragma unroll
    for (int j = 0; j < 8; ++j)
        out[(long long)(rbase + j) * KOUT + colG] = c[j];
}

// ---------------------------------------------------------------------------
// agg[i] = hW[i] * dinv[i]^2  (self-loop contribution doubles as initializer)
// ---------------------------------------------------------------------------
template <int FEAT>
__global__ void selfloop_init_kernel(const float* __restrict__ hW,
                                     const float* __restrict__ dinv,
                                     float* __restrict__ agg, int n) {
    long long tid = (long long)blockIdx.x * blockDim.x + threadIdx.x;
    if (tid >= (long long)n * (FEAT / 4)) return;
    const int node = (int)(tid / (FEAT / 4));
    const float d = dinv[node];
    const float s = d * d;
    float4 v = *reinterpret_cast<const float4*>(hW + tid * 4);
    v.x *= s; v.y *= s; v.z *= s; v.w *= s;
    *reinterpret_cast<float4*>(agg + tid * 4) = v;
}

// ---------------------------------------------------------------------------
// Edge gather-scale-scatter: agg[dst] += hW[src] * norm[e]. FEAT/4 lanes per edge.
// ---------------------------------------------------------------------------
template <int FEAT>
__global__ void edge_scatter_kernel(const int* __restrict__ srcI,
                                    const int* __restrict__ dstI,
                                    const float* __restrict__ nrm,
                                    const float* __restrict__ hW,
                                    float* __restrict__ agg, long long nE) {
    constexpr int TPE = FEAT / 4;
    long long tid = (long long)blockIdx.x * blockDim.x + threadIdx.x;
    long long e = tid / TPE;
    if (e >= nE) return;
    const int f = (int)(tid % TPE) * 4;
    if ((threadIdx.x & 63) == 0) {
        __builtin_prefetch(srcI + e + 2048, 0, 0);
        __builtin_prefetch(nrm + e + 2048, 0, 0);
    }
    const int s = srcI[e];
    const int d = dstI[e];
    const float w = nrm[e];
    float4 v = *reinterpret_cast<const float4*>(hW + (long long)s * FEAT + f);
    float* ap = agg + (long long)d * FEAT + f;
    unsafeAtomicAdd(ap + 0, v.x * w);
    unsafeAtomicAdd(ap + 1, v.y * w);
    unsafeAtomicAdd(ap + 2, v.z * w);
    unsafeAtomicAdd(ap + 3, v.w * w);
}

// ---------------------------------------------------------------------------
// h += bias (broadcast over rows), optional ReLU, in place, float4-vectorized.
// ---------------------------------------------------------------------------
template <int FEAT, bool RELU>
__global__ void bias_act_kernel(float* __restrict__ h, const float* __restrict__ bias, int n) {
    long long tid = (long long)blockIdx.x * blockDim.x + threadIdx.x;
    if (tid >= (long long)n * (FEAT / 4)) return;
    const int f = (int)((tid * 4) % FEAT);
    float4 v = *reinterpret_cast<float4*>(h + tid * 4);
    v.x += bias[f + 0]; v.y += bias[f + 1]; v.z += bias[f + 2]; v.w += bias[f + 3];
    if (RELU) {
        v.x = fmaxf(v.x, 0.0f); v.y = fmaxf(v.y, 0.0f);
        v.z = fmaxf(v.z, 0.0f); v.w = fmaxf(v.w, 0.0f);
    }
    *reinterpret_cast<float4*>(h + tid * 4) = v;
}

// ---------------------------------------------------------------------------
// Global mean pool: scatter-add node features per graph + counts, then divide.
// ---------------------------------------------------------------------------
__global__ void pool_scatter_kernel(const float* __restrict__ h,
                                    const long long* __restrict__ batch,
                                    float* __restrict__ pool, float* __restrict__ cnt,
                                    int n) {
    long long tid = (long long)blockIdx.x * blockDim.x + threadIdx.x;
    const int node = (int)(tid >> 5);
    const int f = (int)(tid & 31);
    if (node >= n) return;
    const int g = (int)batch[node];
    unsafeAtomicAdd(pool + g * 32 + f, h[(long long)node * 32 + f]);
    if (f == 0) unsafeAtomicAdd(cnt + g, 1.0f);
}

__global__ void pool_final_kernel(const float* __restrict__ pool,
                                  const float* __restrict__ cnt,
                                  float* __restrict__ out) {
    int i = blockIdx.x * blockDim.x + threadIdx.x;
    if (i >= N_GRAPHS * 32) return;
    out[i] = pool[i] / fmaxf(cnt[i >> 5], 1.0f);
}

// ---------------------------------------------------------------------------
// Launcher
// ---------------------------------------------------------------------------
extern "C" void kernel_launch(void* const* d_in, const int* in_sizes, int n_in,
                              void* d_out, int out_size, void* d_ws, size_t ws_size,
                              hipStream_t stream) {
    const float*     x      = (const float*)d_in[0];
    const long long* eidx   = (const long long*)d_in[1];   // (2, E) int64
    const long long* batch  = (const long long*)d_in[2];
    const float*     W1     = (const float*)d_in[3];
    const float*     b1     = (const float*)d_in[4];
    const float*     W2     = (const float*)d_in[5];
    const float*     b2     = (const float*)d_in[6];
    const float*     W3     = (const float*)d_in[7];
    const float*     b3     = (const float*)d_in[8];
    float* out = (float*)d_out;

    const long long* src64 = eidx;
    const long long* dst64 = eidx + N_EDGES;

    // Workspace layout (~90 MB, all 4-byte typed)
    float* ws   = (float*)d_ws;
    float* dinv = ws;                                   // N      (degree, then rsqrt in place)
    int*   srcI = (int*)(dinv + N_NODES);               // E
    int*   dstI = srcI + N_EDGES;                       // E
    float* nrm  = (float*)(dstI + N_EDGES);             // E
    float* bufA = nrm + N_EDGES;                        // N*64 (GEMM output)
    float* bufB = bufA + (size_t)N_NODES * 64;          // N*64 (aggregate / next input)
    float* pool = bufB + (size_t)N_NODES * 64;          // 256*32
    float* cnt  = pool + N_GRAPHS * 32;                 // 256

    const int T = 256;
    const int nBlkN  = (N_NODES + T - 1) / T;                       // 391
    const int nBlkE  = (int)((N_EDGES + T - 1) / T);                // 12500
    const int nBlk64 = (int)(((long long)N_NODES * 16 + T - 1) / T);// 6250  (N*64/4)
    const int nBlk32 = (int)(((long long)N_NODES * 8 + T - 1) / T); // 3125  (N*32/4)
    const int nBlkS64 = (int)((N_EDGES * 16 + T - 1) / T);          // 200000
    const int nBlkS32 = (int)((N_EDGES * 8 + T - 1) / T);           // 100000

    // --- normalization: deg -> dinv -> norm (once, reused by all layers) ---
    fill_kernel<<<nBlkN, T, 0, stream>>>(dinv, 1.0f, N_NODES);      // self-loop degree
    edge_prep_kernel<<<nBlkE, T, 0, stream>>>(src64, dst64, srcI, dstI, dinv, N_EDGES);
    rsqrt_kernel<<<nBlkN, T, 0, stream>>>(dinv, N_NODES);
    norm_kernel<<<nBlkE, T, 0, stream>>>(srcI, dstI, dinv, nrm, N_EDGES);

    // --- layer 1: x @ W1 (13->64), aggregate, +b1, ReLU ---
    gemm_wmma_kernel<13, 64><<<(6250 * 4) / 8, T, 0, stream>>>(x, W1, bufA, N_NODES);
    selfloop_init_kernel<64><<<nBlk64, T, 0, stream>>>(bufA, dinv, bufB, N_NODES);
    edge_scatter_kernel<64><<<nBlkS64, T, 0, stream>>>(srcI, dstI, nrm, bufA, bufB, N_EDGES);
    bias_act_kernel<64, true><<<nBlk64, T, 0, stream>>>(bufB, b1, N_NODES);

    // --- layer 2: h @ W2 (64->64), aggregate, +b2, ReLU ---
    gemm_wmma_kernel<64, 64><<<(6250 * 4) / 8, T, 0, stream>>>(bufB, W2, bufA, N_NODES);
    selfloop_init_kernel<64><<<nBlk64, T, 0, stream>>>(bufA, dinv, bufB, N_NODES);
    edge_scatter_kernel<64><<<nBlkS64, T, 0, stream>>>(srcI, dstI, nrm, bufA, bufB, N_EDGES);
    bias_act_kernel<64, true><<<nBlk64, T, 0, stream>>>(bufB, b2, N_NODES);

    // --- layer 3: h @ W3 (64->32), aggregate, +b3 (no ReLU) ---
    gemm_wmma_kernel<64, 32><<<(6250 * 2 + 7) / 8, T, 0, stream>>>(bufB, W3, bufA, N_NODES);
    selfloop_init_kernel<32><<<nBlk32, T, 0, stream>>>(bufA, dinv, bufB, N_NODES);
    edge_scatter_kernel<32><<<nBlkS32, T, 0, stream>>>(srcI, dstI, nrm, bufA, bufB, N_EDGES);
    bias_act_kernel<32, false><<<nBlk32, T, 0, stream>>>(bufB, b3, N_NODES);

    // --- global mean pool ---
    fill_kernel<<<(N_GRAPHS * 32 + N_GRAPHS + T - 1) / T, T, 0, stream>>>(pool, 0.0f,
                                                                          N_GRAPHS * 32 + N_GRAPHS);
    pool_scatter_kernel<<<(int)(((long long)N_NODES * 32 + T - 1) / T), T, 0, stream>>>(
        bufB, batch, pool, cnt, N_NODES);
    pool_final_kernel<<<(N_GRAPHS * 32 + T - 1) / T, T, 0, stream>>>(pool, cnt, out);
}